// _ProposalLayer_7739531067881
// MI455X (gfx1250) — compile-verified
//
#include <hip/hip_runtime.h>
#include <hip/hip_bf16.h>
#include <stdint.h>

#define B_     4
#define A_     9
#define H_     160
#define W_     240
#define N_     (H_*W_*A_)      /* 345600 */
#define PRE_   12000
#define POST_  2000
#define NMS_T  0.7f
#define NSORT_ 16384           /* candidate pool, power of two for bitonic sort */
#define NBUCK_ 16384           /* 14-bit radix buckets */
#define FEAT_  16.0f
#define MASKW_ ((PRE_ + 31) / 32)   /* 375 words * 32 = 12000 exactly, no tail bits */

#if __has_builtin(__builtin_amdgcn_ballot_w32)
#define BALLOT32(p) __builtin_amdgcn_ballot_w32(p)
#else
#define BALLOT32(p) ((uint32_t)__ballot(p))
#endif

// generate_anchors(16, [0.5,1,2], [8,16,32]) evaluated at compile time
__constant__ float ANCH[A_][4] = {
  { -84.f,  -40.f,  99.f,  55.f}, {-176.f,  -88.f, 191.f, 103.f}, {-360.f, -184.f, 375.f, 199.f},
  { -56.f,  -56.f,  71.f,  71.f}, {-120.f, -120.f, 135.f, 135.f}, {-248.f, -248.f, 263.f, 263.f},
  { -36.f,  -80.f,  51.f,  95.f}, { -80.f, -168.f,  95.f, 183.f}, {-168.f, -344.f, 183.f, 359.f}};

__global__ void k_init(uint32_t* __restrict__ hist, uint32_t* __restrict__ cnt,
                       unsigned long long* __restrict__ cand) {
  int t = blockIdx.x * blockDim.x + threadIdx.x;
  int stride = gridDim.x * blockDim.x;
  for (int i = t; i < B_ * NBUCK_; i += stride) hist[i] = 0u;
  for (int i = t; i < B_ * NSORT_; i += stride) cand[i] = ~0ull;
  if (t < B_) cnt[t] = 0u;
}

// Decode anchors+deltas -> clipped proposals; build sortable descending key; bucket histogram.
__global__ void k_props(const float* __restrict__ scores, const float* __restrict__ deltas,
                        const float* __restrict__ iminfo, float4* __restrict__ prop,
                        uint32_t* __restrict__ keys, uint32_t* __restrict__ hist) {
  int gid = blockIdx.x * blockDim.x + threadIdx.x;
  if (gid >= B_ * N_) return;
  int b = gid / N_, n = gid - b * N_;
  int a = n % A_, hw = n / A_;
  int w = hw % W_, h = hw / W_;
  const int HW = H_ * W_;

  float sx = (float)w * FEAT_, sy = (float)h * FEAT_;
  float x1 = ANCH[a][0] + sx, y1 = ANCH[a][1] + sy;
  float x2 = ANCH[a][2] + sx, y2 = ANCH[a][3] + sy;
  float aw = x2 - x1 + 1.f, ah = y2 - y1 + 1.f;
  float cx = x1 + 0.5f * aw, cy = y1 + 0.5f * ah;

  int dbase = ((b * 4 * A_ + a * 4) * H_ + h) * W_ + w;
  __builtin_prefetch(&deltas[dbase + 4 * HW], 0, 1);   // global_prefetch_b8 (streaming hint)
  float dx = deltas[dbase], dy = deltas[dbase + HW];
  float dw = deltas[dbase + 2 * HW], dh = deltas[dbase + 3 * HW];

  float pcx = dx * aw + cx, pcy = dy * ah + cy;
  float pw = expf(dw) * aw, ph = expf(dh) * ah;
  float hmax = iminfo[b * 3 + 0] - 1.f, wmax = iminfo[b * 3 + 1] - 1.f;
  float bx1 = fminf(fmaxf(pcx - 0.5f * pw, 0.f), wmax);
  float by1 = fminf(fmaxf(pcy - 0.5f * ph, 0.f), hmax);
  float bx2 = fminf(fmaxf(pcx + 0.5f * pw, 0.f), wmax);
  float by2 = fminf(fmaxf(pcy + 0.5f * ph, 0.f), hmax);
  prop[gid] = make_float4(bx1, by1, bx2, by2);

  float s = scores[((b * 2 * A_ + A_ + a) * H_ + h) * W_ + w];
  uint32_t u = __float_as_uint(s);
  u = (u & 0x80000000u) ? ~u : (u | 0x80000000u);  // order-preserving ascending
  uint32_t key = ~u;                               // ascending key == descending score
  keys[gid] = key;
  atomicAdd(&hist[b * NBUCK_ + (key >> 18)], 1u);
}

// Per batch: smallest bucket T with cumulative count >= PRE_.
__global__ void k_thresh(const uint32_t* __restrict__ hist, uint32_t* __restrict__ thr) {
  int b = threadIdx.x;
  if (b >= B_) return;
  uint32_t cum = 0, T = NBUCK_ - 1;
  for (int k = 0; k < NBUCK_; ++k) {
    cum += hist[b * NBUCK_ + k];
    if (cum >= PRE_) { T = (uint32_t)k; break; }
  }
  thr[b] = T;
}

__global__ void k_compact(const uint32_t* __restrict__ keys, const uint32_t* __restrict__ thr,
                          uint32_t* __restrict__ cnt, unsigned long long* __restrict__ cand) {
  int gid = blockIdx.x * blockDim.x + threadIdx.x;
  if (gid >= B_ * N_) return;
  int b = gid / N_;
  uint32_t n = (uint32_t)(gid - b * N_);
  uint32_t key = keys[gid];
  if ((key >> 18) <= thr[b]) {
    uint32_t pos = atomicAdd(&cnt[b], 1u);
    if (pos < NSORT_) cand[b * NSORT_ + pos] = ((unsigned long long)key << 32) | n;
  }
}

// One workgroup per batch: 16384 x u64 bitonic sort in 128KB LDS (CDNA5 big-LDS).
// Candidates staged into LDS with async global->LDS b128 copies.
__global__ __launch_bounds__(1024) void k_sort(const unsigned long long* __restrict__ cand,
                                               uint32_t* __restrict__ order) {
  extern __shared__ unsigned long long sk[];
  const int b = blockIdx.x, tid = threadIdx.x;
  const unsigned long long* src = cand + (size_t)b * NSORT_;

  // Async copy: 16B per lane per issue (2 x u64), 8 issues per lane.
  for (int i = tid; i < NSORT_ / 2; i += 1024) {
    unsigned long long g = (unsigned long long)(uintptr_t)(src + 2 * i);
    uint32_t l = (uint32_t)(uintptr_t)(sk + 2 * i);   // generic LDS ptr low 32 bits = LDS offset
    asm volatile("global_load_async_to_lds_b128 %0, %1, off"
                 :: "v"(l), "v"(g) : "memory");
  }
  asm volatile("s_wait_asynccnt 0" ::: "memory");
  __syncthreads();

  for (unsigned k = 2; k <= NSORT_; k <<= 1) {
    for (unsigned j = k >> 1; j > 0; j >>= 1) {
      for (unsigned i = tid; i < NSORT_; i += 1024) {
        unsigned p = i ^ j;
        if (p > i) {
          unsigned long long va = sk[i], vb = sk[p];
          bool up = ((i & k) == 0);
          if (up ? (va > vb) : (va < vb)) { sk[i] = vb; sk[p] = va; }
        }
      }
      __syncthreads();
    }
  }
  for (int i = tid; i < PRE_; i += 1024)
    order[b * PRE_ + i] = (uint32_t)(sk[i] & 0xFFFFFFFFull);
}

// One workgroup per batch: greedy NMS with the full 12000-box set resident in LDS
// (192KB boxes + 48KB areas + bitmask + scan buffer ~ 246KB < 320KB WGP LDS).
// Boxes gathered into LDS via per-lane async global->LDS b128.
// Suppression via wave32 ballot: one wave owns one 32-bit mask word -> no LDS atomics.
__global__ __launch_bounds__(1024) void k_nms(const float4* __restrict__ prop,
                                              const uint32_t* __restrict__ order,
                                              float* __restrict__ out) {
  extern __shared__ char smem[];
  float4*   bx   = (float4*)smem;                                     // PRE_*16
  float*    ar   = (float*)(smem + (size_t)PRE_ * 16);                // PRE_*4
  uint32_t* mask = (uint32_t*)(smem + (size_t)PRE_ * 20);             // MASKW_*4
  uint32_t* scan = (uint32_t*)(smem + (size_t)PRE_ * 20 + ((MASKW_ + 1) & ~1) * 4);
  const int b = blockIdx.x, tid = threadIdx.x;
  const int warp = tid >> 5, lane = tid & 31;

  // Default output rows: [b, 0,0,0,0]
  for (int t = tid; t < POST_; t += 1024) {
    float* row = out + (size_t)(b * POST_ + t) * 5;
    row[0] = (float)b; row[1] = 0.f; row[2] = 0.f; row[3] = 0.f; row[4] = 0.f;
  }
  for (int t = tid; t < MASKW_; t += 1024) mask[t] = 0u;

  // Per-lane async gather of 16B boxes straight into LDS.
  for (int k = tid; k < PRE_; k += 1024) {
    uint32_t idx = order[b * PRE_ + k];
    unsigned long long g = (unsigned long long)(uintptr_t)(prop + (size_t)b * N_ + idx);
    uint32_t l = (uint32_t)(uintptr_t)(bx + k);
    asm volatile("global_load_async_to_lds_b128 %0, %1, off"
                 :: "v"(l), "v"(g) : "memory");
  }
  asm volatile("s_wait_asynccnt 0" ::: "memory");
  __syncthreads();
  for (int k = tid; k < PRE_; k += 1024) {
    float4 p = bx[k];
    ar[k] = (p.z - p.x + 1.f) * (p.w - p.y + 1.f);
  }
  __syncthreads();

  // Greedy NMS: barrier only after kept rows (suppressed rows write nothing).
  for (int i = 0; i < PRE_; ++i) {
    if (!((mask[i >> 5] >> (i & 31)) & 1u)) {
      float4 bi = bx[i];
      float  ai = ar[i];
      int w0 = (i + 1) >> 5;
      for (int wd = w0 + warp; wd < MASKW_; wd += 32) {   // 32 waves, one word per wave
        uint32_t cur = mask[wd];                          // uniform within the wave
        if (cur == 0xFFFFFFFFu) continue;                 // whole word already suppressed
        int j = (wd << 5) + lane;
        bool pred = false;
        if (j > i && !((cur >> lane) & 1u)) {
          float4 bj = bx[j];
          float xx1 = fmaxf(bi.x, bj.x), yy1 = fmaxf(bi.y, bj.y);
          float xx2 = fminf(bi.z, bj.z), yy2 = fminf(bi.w, bj.w);
          float iw = fmaxf(0.f, xx2 - xx1 + 1.f);
          float ih = fmaxf(0.f, yy2 - yy1 + 1.f);
          float inter = iw * ih;
          pred = inter > NMS_T * (ai + ar[j] - inter);    // iou > t, division-free
        }
        uint32_t m = BALLOT32(pred);
        if (lane == 0 && m) mask[wd] |= m;                // word exclusively owned this pass
      }
      __syncthreads();
    }
  }

  // Prefix-scan compaction of survivors, first POST_ written out.
  const int CH = (PRE_ + 1023) / 1024;   // 12 keys per thread, contiguous chunks
  int k0 = tid * CH;
  int kend = k0 + CH; if (kend > PRE_) kend = PRE_;
  uint32_t lc = 0;
  for (int k = k0; k < kend; ++k) lc += !((mask[k >> 5] >> (k & 31)) & 1u);
  scan[tid] = lc;
  __syncthreads();
  for (int d = 1; d < 1024; d <<= 1) {
    uint32_t add = (tid >= d) ? scan[tid - d] : 0u;
    __syncthreads();
    scan[tid] += add;
    __syncthreads();
  }
  uint32_t pos = (tid == 0) ? 0u : scan[tid - 1];
  for (int k = k0; k < kend; ++k) {
    if (!((mask[k >> 5] >> (k & 31)) & 1u)) {
      if (pos < POST_) {
        float4 p = bx[k];
        float* row = out + (size_t)(b * POST_ + pos) * 5;
        row[1] = p.x; row[2] = p.y; row[3] = p.z; row[4] = p.w;
      }
      ++pos;
    }
  }
}

extern "C" void kernel_launch(void* const* d_in, const int* in_sizes, int n_in,
                              void* d_out, int out_size, void* d_ws, size_t ws_size,
                              hipStream_t stream) {
  const float* scores = (const float*)d_in[0];
  const float* deltas = (const float*)d_in[1];
  const float* iminfo = (const float*)d_in[2];
  float* out = (float*)d_out;

  char* ws = (char*)d_ws;
  size_t off = 0;
  auto take = [&](size_t bytes) { size_t o = off; off = (off + bytes + 255) & ~(size_t)255; return o; };
  float4*             prop  = (float4*)(ws + take((size_t)B_ * N_ * 16));
  uint32_t*           keys  = (uint32_t*)(ws + take((size_t)B_ * N_ * 4));
  uint32_t*           hist  = (uint32_t*)(ws + take((size_t)B_ * NBUCK_ * 4));
  uint32_t*           thr   = (uint32_t*)(ws + take((size_t)B_ * 4));
  uint32_t*           cnt   = (uint32_t*)(ws + take((size_t)B_ * 4));
  unsigned long long* cand  = (unsigned long long*)(ws + take((size_t)B_ * NSORT_ * 8));
  uint32_t*           order = (uint32_t*)(ws + take((size_t)B_ * PRE_ * 4));

  k_init<<<256, 256, 0, stream>>>(hist, cnt, cand);

  int nb = (B_ * N_ + 255) / 256;
  k_props<<<nb, 256, 0, stream>>>(scores, deltas, iminfo, prop, keys, hist);
  k_thresh<<<1, 32, 0, stream>>>(hist, thr);
  k_compact<<<nb, 256, 0, stream>>>(keys, thr, cnt, cand);

  k_sort<<<B_, 1024, (size_t)NSORT_ * 8, stream>>>(cand, order);

  size_t nms_lds = (size_t)PRE_ * 20 + ((MASKW_ + 1) & ~1) * 4 + 1024 * 4;
  k_nms<<<B_, 1024, nms_lds, stream>>>(prop, order, out);
}